// KPRE_5248450035741
// MI455X (gfx1250) — compile-verified
//
#include <hip/hip_runtime.h>
#include <math.h>

namespace {
constexpr int B = 1024, K = 50, P = 512, L = 3, D = 64, NREL = 32;
constexpr int KPAD = 64;           // neighbor rows padded to 4 WMMA tiles
constexpr int UISTR = 132;         // padded LDS row stride (floats) to dodge bank conflicts
constexpr int THREADS = 256;
}

typedef float v2f __attribute__((ext_vector_type(2)));
typedef float v8f __attribute__((ext_vector_type(8)));

__device__ __forceinline__ float wave_sum(float v) {
#pragma unroll
  for (int o = 16; o > 0; o >>= 1) v += __shfl_xor(v, o, 32);
  return v;
}
__device__ __forceinline__ float wave_max(float v) {
#pragma unroll
  for (int o = 16; o > 0; o >>= 1) v = fmaxf(v, __shfl_xor(v, o, 32));
  return v;
}

__global__ __launch_bounds__(THREADS) void kpre_kernel(
    const int* __restrict__ users, const int* __restrict__ items,
    const int* __restrict__ ut_users, const int* __restrict__ ut_items,
    const int* __restrict__ it_heads, const int* __restrict__ it_rels,
    const int* __restrict__ it_tails,
    const float* __restrict__ user_emb, const float* __restrict__ entity_emb,
    const float* __restrict__ rel_emb, const float* __restrict__ att_w,
    float* __restrict__ out)
{
  __shared__ __align__(16) float relLds[NREL * D];   // 8 KB relation table
  __shared__ __align__(16) float ui[KPAD][UISTR];    // [ue|ie] neighbor tile, ~33 KB
  __shared__ __align__(16) float4 accLds[16][16];    // path partial sums, 4 KB
  __shared__ __align__(16) float attw[2 * D];
  __shared__ __align__(16) float pav[D];
  __shared__ __align__(16) float lg[KPAD];

  const int b = blockIdx.x;
  const int tid = threadIdx.x;

  // ---- stage relation table (32x64) + attention weights into LDS ----
  for (int i = tid; i < NREL * D / 4; i += THREADS)
    ((float4*)relLds)[i] = ((const float4*)rel_emb)[i];
  if (tid < 2 * D) attw[tid] = att_w[tid];

  // ---- stage [ue|ie] tile: KPAD rows x 128 cols, zero-padded past K ----
  {
    const int row0 = tid >> 4;   // 16 concurrent rows
    const int g = tid & 15;      // float4 group within 64 floats (coalesced 256B rows)
    for (int k = row0; k < KPAD; k += 16) {
      float4 u4 = make_float4(0.f, 0.f, 0.f, 0.f);
      float4 e4 = make_float4(0.f, 0.f, 0.f, 0.f);
      if (k < K) {
        const int uu = ut_users[b * K + k];
        const int ii = ut_items[b * K + k];
        u4 = ((const float4*)(user_emb + (size_t)uu * D))[g];
        e4 = ((const float4*)(entity_emb + (size_t)ii * D))[g];
      }
      *(float4*)&ui[k][g * 4] = u4;
      *(float4*)&ui[k][D + g * 4] = e4;
    }
  }
  __syncthreads();

  // ---- phase 1: path aggregation (memory-bound gather core) ----
  const int dg = tid & 15;     // dim group (float4)
  const int slot = tid >> 4;   // path slot
  float ax = 0.f, ay = 0.f, az = 0.f, aw = 0.f;
  for (int p = slot; p < P; p += 16) {
    const int head = it_heads[b * P + p];
    const int r0 = it_rels[(b * L + 0) * P + p];
    const int r1 = it_rels[(b * L + 1) * P + p];
    const int r2 = it_rels[(b * L + 2) * P + p];
    const int t0 = it_tails[(b * L + 0) * P + p];
    const int t1 = it_tails[(b * L + 1) * P + p];
    const int t2 = it_tails[(b * L + 2) * P + p];

    float4 e = ((const float4*)(entity_emb + (size_t)t2 * D))[dg];
    float4 r = *(const float4*)&relLds[r2 * D + dg * 4];
    float wx = r.x * e.x, wy = r.y * e.y, wz = r.z * e.z, ww = r.w * e.w;

    e = ((const float4*)(entity_emb + (size_t)t1 * D))[dg];
    r = *(const float4*)&relLds[r1 * D + dg * 4];
    wx = (wx + e.x) * r.x; wy = (wy + e.y) * r.y;
    wz = (wz + e.z) * r.z; ww = (ww + e.w) * r.w;

    e = ((const float4*)(entity_emb + (size_t)t0 * D))[dg];
    r = *(const float4*)&relLds[r0 * D + dg * 4];
    wx = (wx + e.x) * r.x; wy = (wy + e.y) * r.y;
    wz = (wz + e.z) * r.z; ww = (ww + e.w) * r.w;

    e = ((const float4*)(entity_emb + (size_t)head * D))[dg];
    ax += wx + e.x; ay += wy + e.y; az += wz + e.z; aw += ww + e.w;
  }
  accLds[slot][dg] = make_float4(ax, ay, az, aw);
  __syncthreads();
  if (tid < 16) {
    float sx = 0.f, sy = 0.f, sz = 0.f, sw = 0.f;
    for (int s = 0; s < 16; ++s) {
      float4 v = accLds[s][tid];
      sx += v.x; sy += v.y; sz += v.z; sw += v.w;
    }
    const float sc = 1.0f / ((float)(L + 1) * (float)P);
    pav[tid * 4 + 0] = sx * sc; pav[tid * 4 + 1] = sy * sc;
    pav[tid * 4 + 2] = sz * sc; pav[tid * 4 + 3] = sw * sc;
  }
  __syncthreads();

  // ---- phase 2a: attention logits via V_WMMA_F32_16X16X4_F32 (wave 0) ----
  // D[16x16] = A[16x4] * B[4x16] + C; A rows = neighbors, B = att_w chunk
  // broadcast across all 16 columns, accumulate over 32 K-chunks.
  if (tid < 32) {
    const int l = tid;
    const int half = (l >> 4) << 1;   // lanes 0-15 carry K=0,1; lanes 16-31 carry K=2,3
    const int lrow = l & 15;
#pragma unroll
    for (int t = 0; t < 4; ++t) {
      const int row = t * 16 + lrow;
      v8f c = {0.f, 0.f, 0.f, 0.f, 0.f, 0.f, 0.f, 0.f};
#pragma unroll
      for (int ch = 0; ch < 32; ++ch) {
        const int k0 = ch * 4 + half;
        const float2 av = *(const float2*)&ui[row][k0];
        const float2 bv = *(const float2*)&attw[k0];
        v2f a; a.x = av.x; a.y = av.y;
        v2f bb; bb.x = bv.x; bb.y = bv.y;
        c = __builtin_amdgcn_wmma_f32_16x16x4_f32(false, a, false, bb,
                                                  (short)0, c, false, false);
      }
      // column 0 of D: lane 0 holds rows 0-7 (VGPR 0-7), lane 16 rows 8-15
      if (l == 0 || l == 16) {
        const int base = t * 16 + ((l == 16) ? 8 : 0);
#pragma unroll
        for (int rI = 0; rI < 8; ++rI) {
          const float wl = c[rI];
          lg[base + rI] = (wl > 0.f) ? wl : 0.01f * wl;   // leaky_relu(0.01)
        }
      }
    }
  }
  __syncthreads();

  // ---- phase 2b: softmax + weighted agg + final score (wave 0) ----
  if (tid < 32) {
    const int l = tid;
    const float v0 = (l < K) ? lg[l] : -3.0e38f;
    const float v1 = (l + 32 < K) ? lg[l + 32] : -3.0e38f;
    const float m = wave_max(fmaxf(v0, v1));
    const float e0 = (l < K) ? __expf(v0 - m) : 0.f;
    const float e1 = (l + 32 < K) ? __expf(v1 - m) : 0.f;
    const float inv = 1.0f / wave_sum(e0 + e1);

    float a0 = 0.f, a1 = 0.f;   // item_agg dims l and l+32, kept in registers
    for (int k = 0; k < K; ++k) {
      const float sk = (k < 32) ? __shfl(e0, k, 32) : __shfl(e1, k - 32, 32);
      a0 += ui[k][D + l] * sk;
      a1 += ui[k][D + l + 32] * sk;
    }
    a0 *= inv; a1 *= inv;

    const int u = users[b];
    const int it = items[b];
    const float* ue = user_emb + (size_t)u * D;
    const float* ie = entity_emb + (size_t)it * D;
    float part = ue[l] * ie[l] + ue[l + 32] * ie[l + 32] +
                 a0 * pav[l] + a1 * pav[l + 32];
    part = wave_sum(part);
    if (l == 0) out[b] = 1.0f / (1.0f + __expf(-part));
  }
}

extern "C" void kernel_launch(void* const* d_in, const int* in_sizes, int n_in,
                              void* d_out, int out_size, void* d_ws, size_t ws_size,
                              hipStream_t stream) {
  (void)in_sizes; (void)n_in; (void)out_size; (void)d_ws; (void)ws_size;
  const int* users      = (const int*)d_in[0];
  const int* items      = (const int*)d_in[1];
  const int* ut_users   = (const int*)d_in[2];
  const int* ut_items   = (const int*)d_in[3];
  const int* it_heads   = (const int*)d_in[4];
  const int* it_rels    = (const int*)d_in[5];
  const int* it_tails   = (const int*)d_in[6];
  const float* user_emb   = (const float*)d_in[7];
  const float* entity_emb = (const float*)d_in[8];
  const float* rel_emb    = (const float*)d_in[9];
  const float* att_w      = (const float*)d_in[10];
  float* out = (float*)d_out;

  kpre_kernel<<<B, THREADS, 0, stream>>>(users, items, ut_users, ut_items,
                                         it_heads, it_rels, it_tails,
                                         user_emb, entity_emb, rel_emb, att_w,
                                         out);
}